// CTCLayer_27702539059403
// MI455X (gfx1250) — compile-verified
//
#include <hip/hip_runtime.h>
#include <cstdint>

// CTC batch cost, MI455X (gfx1250), wave32.
// B=512, T=256, C=256, L=64, S=2L+1=129. Output: (B,1) float32.
#define B_  512
#define T_  256
#define C_  256
#define L_  64
#define S_  129
#define NBUF 4                 // circular row-buffer depth (power of 2)
#define NEGV (-1e30f)
#define EPSV (1e-7f)

// Low 32 bits of a flat pointer into LDS == workgroup-relative LDS byte offset
// (shared-aperture flat addresses carry the LDS offset in addr[31:0]).
__device__ __forceinline__ uint32_t lds_off32(const void* p) {
  return (uint32_t)(uintptr_t)p;
}

// Per-lane async copy: 8 bytes global -> LDS, tracked by ASYNCcnt (CDNA5 path).
// GV addressing: 64-bit global address in a VGPR pair, SADDR = off.
__device__ __forceinline__ void async_ld_b64(const float* gsrc, float* ldst) {
  uint64_t ga = (uint64_t)(uintptr_t)gsrc;
  uint32_t la = lds_off32(ldst);
  asm volatile("global_load_async_to_lds_b64 %0, %1, off"
               :: "v"(la), "v"(ga) : "memory");
}

__device__ __forceinline__ void wait_async_le(int k) {  // k uniform
  if (k >= 2)      asm volatile("s_wait_asynccnt 0x2" ::: "memory");
  else if (k == 1) asm volatile("s_wait_asynccnt 0x1" ::: "memory");
  else             asm volatile("s_wait_asynccnt 0x0" ::: "memory");
}

__global__ __launch_bounds__(160) void ctc_cost_kernel(
    const float* __restrict__ y_pred,        // (B,T,C)
    const int*   __restrict__ labels,        // (B,L)
    const int*   __restrict__ input_length,  // (B,1)
    const int*   __restrict__ label_length,  // (B,1)
    float*       __restrict__ out)           // (B,1)
{
  __shared__ __align__(16) float buf[NBUF][C_];  // 4-deep async-staged rows
  __shared__ __align__(16) float alphaA[S_ + 3]; // DP state ping
  __shared__ __align__(16) float alphaB[S_ + 3]; // DP state pong

  const int b   = blockIdx.x;
  const int tid = threadIdx.x;
  const int s   = tid;                 // state owned by this thread (if < S_)

  int il = input_length[b];            // uniform -> scalar load / branches
  il = il < 1 ? 1 : (il > T_ ? T_ : il);

  // Extended-label class + skip-allow flag. Even s -> blank (C-1); odd s ->
  // labels[b][s>>1] (never blank: randint high bound C-1 exclusive), so
  // allow_skip = odd && s>=3 && lab[k] != lab[k-1].
  int  ext   = C_ - 1;
  bool allow = false;
  if (s < S_ && (s & 1)) {
    const int k = s >> 1;
    ext = labels[b * L_ + k];
    if (s >= 3) allow = (ext != labels[b * L_ + k - 1]);
  }

  const float* rowbase = y_pred + (size_t)b * T_ * C_;
  const bool loader = (tid < 128);     // waves 0..3, full EXEC per wave

  // ---- Preload rows 0..3 (async; retire in order per wave) ----
  if (loader) {
    const int e = tid << 1;            // 2 floats per lane, 128 lanes = 1 row
    async_ld_b64(rowbase + 0 * C_ + e, &buf[0][e]);
    async_ld_b64(rowbase + 1 * C_ + e, &buf[1][e]);
    async_ld_b64(rowbase + 2 * C_ + e, &buf[2][e]);
    async_ld_b64(rowbase + 3 * C_ + e, &buf[3][e]);
  }
  asm volatile("s_wait_asynccnt 0x3" ::: "memory");   // row 0 landed
  __syncthreads();

  float* aCur = alphaA;
  float* aNxt = alphaB;

  // t = 0: alpha0[0] = emit(blank), alpha0[1] = emit(label0), else NEG
  if (s < S_) {
    float a0 = NEGV;
    if (s <= 1) a0 = __logf(buf[0][ext] + EPSV);
    aCur[s] = a0;
  }

  // ---- DP scan over active timesteps only (t >= il leaves alpha fixed) ----
  // One barrier per step: it simultaneously (a) publishes row t + alpha(t-1),
  // (b) proves all waves finished reading buf[(t-1)&3] and the old alpha, so
  // the prefetch of row t+3 into buf[(t+3)&3]==buf[(t-1)&3] and the write of
  // aNxt (read last at t-1) are both safe immediately after it.
  for (int t = 1; t < il; ++t) {
    wait_async_le(min(2, il - 1 - t)); // retire through row t, keep rest in flight
    __syncthreads();

    if (loader && (t + 3) < il) {      // refill slot last read at t-1
      const int e = tid << 1;
      async_ld_b64(rowbase + (size_t)(t + 3) * C_ + e, &buf[(t + 3) & (NBUF - 1)][e]);
    }

    if (s < S_) {
      const float* row = buf[t & (NBUF - 1)];
      const float emit = __logf(row[ext] + EPSV);      // LDS gather
      const float a  = aCur[s];
      const float a1 = (s >= 1) ? aCur[s - 1] : NEGV;
      const float a2 = allow    ? aCur[s - 2] : NEGV;
      const float m  = fmaxf(a, fmaxf(a1, a2));
      const float lse = m + __logf(__expf(a - m) + __expf(a1 - m) + __expf(a2 - m));
      aNxt[s] = lse + emit;
    }

    float* tmp = aCur; aCur = aNxt; aNxt = tmp;
  }

  __syncthreads();                     // publish final alpha
  if (tid == 0) {
    int ll = label_length[b];
    ll = ll < 1 ? 1 : (ll > L_ ? L_ : ll);
    const int i1 = 2 * ll - 1, i2 = 2 * ll;            // i2 <= 2L < S_
    const float x = aCur[i1], y = aCur[i2];
    const float m = fmaxf(x, y);
    out[b] = -(m + __logf(__expf(x - m) + __expf(y - m)));
  }
}

extern "C" void kernel_launch(void* const* d_in, const int* in_sizes, int n_in,
                              void* d_out, int out_size, void* d_ws, size_t ws_size,
                              hipStream_t stream) {
  (void)in_sizes; (void)n_in; (void)d_ws; (void)ws_size; (void)out_size;
  const float* y_pred       = (const float*)d_in[0];
  const int*   labels       = (const int*)d_in[1];
  const int*   input_length = (const int*)d_in[2];
  const int*   label_length = (const int*)d_in[3];
  float*       out          = (float*)d_out;

  ctc_cost_kernel<<<dim3(B_), dim3(160), 0, stream>>>(
      y_pred, labels, input_length, label_length, out);
}